// MeanShiftClustering_60481729462634
// MI455X (gfx1250) — compile-verified
//
#include <hip/hip_runtime.h>
#include <hip/hip_bf16.h>
#include <climits>

// MeanShift clustering refinement for MI455X (gfx1250, wave32, WMMA).
//
// Shapes: embeddings [B=4, C=16, 16,16,16] -> pts [B, N=4096, C=16]
//         seg_logits [B, K=8, 16,16,16]    -> seg [B, K, N]
// Core cost: 20 iterations x 2 GEMMs (4096x4096x16 and 4096x16x4096) in fp32
// per batch. Working set = 256KB/batch -> fully L2 resident; compute bound on
// fp32 WMMA (V_WMMA_F32_16X16X4_F32, exact fp32: the d2<=bw^2 threshold is
// precision sensitive). Point tiles are staged once per block into LDS via the
// CDNA5 async-to-LDS engine (double buffered, ASYNCcnt), removing the 4x
// per-wave VMEM amplification.

typedef __attribute__((ext_vector_type(2))) float v2f;
typedef __attribute__((ext_vector_type(8))) float v8f;

// pointee type required by __builtin_amdgcn_global_load_async_to_lds_b128
// (from the clang diagnostic: vector_size(4*sizeof(int)) int)
typedef __attribute__((vector_size(16))) int b128_t;
typedef __attribute__((address_space(1))) b128_t gb128_t;
typedef __attribute__((address_space(3))) b128_t lb128_t;

#define NB 4
#define NPTS 4096
#define NC 16
#define NK 8
#define BW2 4.0f
#define ITERS 20

#define PTS_LD 20  // padded LDS row stride (floats): 80B, 16B-aligned, bank-spread
#define W_LD 17    // padded LDS row stride for the w tile

// ---- CDNA5 async load-to-LDS ----------------------------------------------
#if defined(__has_builtin)
#if __has_builtin(__builtin_amdgcn_global_load_async_to_lds_b128)
#define HAVE_ASYNC_LDS 1
#endif
#if __has_builtin(__builtin_amdgcn_s_wait_asynccnt)
#define HAVE_WAIT_ASYNC 1
#endif
#endif
#ifndef HAVE_ASYNC_LDS
#define HAVE_ASYNC_LDS 0
#endif
#ifndef HAVE_WAIT_ASYNC
#define HAVE_WAIT_ASYNC 0
#endif

__device__ __forceinline__ void tile_copy16B(const float* g, float* l) {
#if HAVE_ASYNC_LDS
  __builtin_amdgcn_global_load_async_to_lds_b128((gb128_t*)g, (lb128_t*)l, 0, 0);
#else
  *(float4*)l = *(const float4*)g;
#endif
}

__device__ __forceinline__ void tile_copy_wait() {
#if HAVE_ASYNC_LDS
#if HAVE_WAIT_ASYNC
  __builtin_amdgcn_s_wait_asynccnt(0);
#else
  asm volatile("s_wait_asynccnt 0x0" ::: "memory");
#endif
#endif
}

// Stage one 16x16 f32 tile (rows jBase..jBase+15 of a row-major [N,16] array)
// into LDS with padded row stride PTS_LD. Called by wave 0 only; 2x B128/lane.
__device__ __forceinline__ void stage_tile(const float* src16, int jBase,
                                           float* dst, int lane) {
  const float* g = src16 + (size_t)jBase * NC + lane * 4;
  float* l = dst + (lane >> 2) * PTS_LD + (lane & 3) * 4;
  tile_copy16B(g, l);                     // rows 0..7 of the tile
  tile_copy16B(g + 128, l + 8 * PTS_LD);  // rows 8..15
}

// ---------------------------------------------------------------------------
// Kernel 1: transpose emb [C,N] -> ptsRM [N,C] (row-major), init modes, point
// norms, and per-point argmax class of seg_logits.
// ---------------------------------------------------------------------------
__global__ __launch_bounds__(256) void k_prep(const float* __restrict__ emb,
                                              const float* __restrict__ seg,
                                              float* __restrict__ ptsRM,
                                              float* __restrict__ modesA,
                                              float* __restrict__ ptsNorm,
                                              int* __restrict__ segPred) {
  const int tid = blockIdx.x * blockDim.x + threadIdx.x;  // 0..B*N-1
  const int b = tid >> 12;
  const int n = tid & (NPTS - 1);
  const float* e = emb + (size_t)b * NC * NPTS;
  float s = 0.0f;
#pragma unroll
  for (int c = 0; c < NC; ++c) {
    const float v = e[c * NPTS + n];  // coalesced over n
    ptsRM[(size_t)tid * NC + c] = v;
    modesA[(size_t)tid * NC + c] = v;
    s += v * v;
  }
  ptsNorm[tid] = s;

  const float* sg = seg + (size_t)b * NK * NPTS;
  int bk = 0;
  float bv = sg[n];
#pragma unroll
  for (int k = 1; k < NK; ++k) {
    const float v = sg[k * NPTS + n];
    if (v > bv) { bv = v; bk = k; }  // first-max tie break
  }
  segPred[tid] = bk;
}

// ---------------------------------------------------------------------------
// Kernel 2: one flat-kernel mean-shift step.
// One wave = one 16-row mode tile; 4 waves/block share the batch and the
// column sweep, so point tiles are staged once per block into double-buffered
// LDS by wave 0 via async-to-LDS, overlapped with WMMA on the current buffer.
// GEMM1 (d2):      A = modes tile, B = pts^T slice  (4x V_WMMA_F32_16X16X4_F32)
// GEMM2 (w @ pts): A = w tile (wave-private LDS relayout; DS ops are in-order
//                  within a wave so no barrier needed), B = pts tile from LDS.
// ---------------------------------------------------------------------------
__global__ __launch_bounds__(128) void k_meanshift(const float* __restrict__ modesIn,
                                                   float* __restrict__ modesOut,
                                                   const float* __restrict__ ptsRM,
                                                   const float* __restrict__ ptsNorm) {
  __shared__ float ptsT[2][16 * PTS_LD];  // double-buffered shared point tile
  __shared__ float ldsW[4][16 * W_LD];    // per-wave w tile
  const int wave = threadIdx.x >> 5;
  const int lane = threadIdx.x & 31;
  const int hi = lane >> 4;  // high half of wave32
  const int l16 = lane & 15;

  const int tile = blockIdx.x * 4 + wave;  // 0..1023 (same batch across block)
  const int b = tile >> 8;
  const int iBase = (tile & 255) << 4;

  const float* mIn = modesIn + (size_t)b * NPTS * NC;
  float* mOut = modesOut + (size_t)b * NPTS * NC;
  const float* pts = ptsRM + (size_t)b * NPTS * NC;
  const float* pn = ptsNorm + (size_t)b * NPTS;

  // A fragments (modes rows): lane holds row l16; VGPR pair covers k,k+1 with
  // +2 offset for the high half (f32 16x4 A layout).
  v2f a[4];
  float part = 0.0f;
#pragma unroll
  for (int s = 0; s < 4; ++s) {
    const int k = 4 * s + 2 * hi;
    a[s] = *(const v2f*)(mIn + (size_t)(iBase + l16) * NC + k);
    part += a[s].x * a[s].x + a[s].y * a[s].y;
  }
  // full row norm: halves hold complementary k's -> xor-16 combine, then
  // broadcast per C/D-layout row via bpermute
  const float mnFull = part + __shfl_xor(part, 16, 32);
  float mnr[8];
#pragma unroll
  for (int r = 0; r < 8; ++r) mnr[r] = __shfl(mnFull, r + 8 * hi, 32);

  v8f macc = {};
  float cnt[8];
#pragma unroll
  for (int r = 0; r < 8; ++r) cnt[r] = 0.0f;

  // preload tile 0
  if (wave == 0) {
    stage_tile(pts, 0, ptsT[0], lane);
    tile_copy_wait();
  }
  __syncthreads();

  for (int jt = 0; jt < NPTS / 16; ++jt) {
    const int buf = jt & 1;
    const int jBase = jt << 4;
    if (wave == 0 && jt + 1 < NPTS / 16)
      stage_tile(pts, jBase + 16, ptsT[buf ^ 1], lane);  // async fill next

    const float pnL = pn[jBase + l16];

    // GEMM1: acc = modes @ pts^T  (B[k][n] = pts[jBase+n][k], from LDS)
    v8f acc = {};
#pragma unroll
    for (int s = 0; s < 4; ++s) {
      const int k = 4 * s + 2 * hi;
      const v2f bf = *(const v2f*)(&ptsT[buf][l16 * PTS_LD + k]);
      acc = __builtin_amdgcn_wmma_f32_16x16x4_f32(false, a[s], false, bf,
                                                  (short)0, acc, false, false);
    }

    // threshold -> w tile in wave-private LDS (C/D layout -> row-major)
#pragma unroll
    for (int r = 0; r < 8; ++r) {
      const float d2 = mnr[r] + pnL - 2.0f * acc[r];
      const float w = (d2 <= BW2) ? 1.0f : 0.0f;
      cnt[r] += w;
      ldsW[wave][(r + 8 * hi) * W_LD + l16] = w;
    }

    // GEMM2: macc += w @ pts (A2 relayout from LDS; DS in-order within wave)
#pragma unroll
    for (int s = 0; s < 4; ++s) {
      const int k = 4 * s + 2 * hi;
      v2f a2;
      a2.x = ldsW[wave][l16 * W_LD + k];
      a2.y = ldsW[wave][l16 * W_LD + k + 1];
      v2f b2;
      b2.x = ptsT[buf][k * PTS_LD + l16];
      b2.y = ptsT[buf][(k + 1) * PTS_LD + l16];
      macc = __builtin_amdgcn_wmma_f32_16x16x4_f32(false, a2, false, b2,
                                                   (short)0, macc, false, false);
    }

    if (wave == 0) tile_copy_wait();  // next buffer resident
    __syncthreads();                  // all waves done with buf; next ready
  }

  // denominators: sum w over the 16 lanes of each half, then write new modes
#pragma unroll
  for (int r = 0; r < 8; ++r) {
    float c = cnt[r];
    c += __shfl_xor(c, 1, 32);
    c += __shfl_xor(c, 2, 32);
    c += __shfl_xor(c, 4, 32);
    c += __shfl_xor(c, 8, 32);
    const float denom = fmaxf(c, 1.0f);
    mOut[(size_t)(iBase + r + 8 * hi) * NC + l16] = macc[r] / denom;
  }
}

// ---------------------------------------------------------------------------
// Kernel 3: mode row norms (for the label pass).
// ---------------------------------------------------------------------------
__global__ __launch_bounds__(256) void k_norms(const float* __restrict__ modes,
                                               float* __restrict__ modeNorm) {
  const int tid = blockIdx.x * blockDim.x + threadIdx.x;  // 0..B*N-1
  float s = 0.0f;
#pragma unroll
  for (int c = 0; c < NC; ++c) {
    const float v = modes[(size_t)tid * NC + c];
    s += v * v;
  }
  modeNorm[tid] = s;
}

// ---------------------------------------------------------------------------
// Kernel 4: labels[i] = first j with ||m_i - m_j||^2 <= bw^2  (== min such j).
// Same WMMA distance tiling, modes vs modes, async double-buffered B tiles.
// ---------------------------------------------------------------------------
__global__ __launch_bounds__(128) void k_labels(const float* __restrict__ modes,
                                                const float* __restrict__ modeNorm,
                                                int* __restrict__ labels) {
  __shared__ float mT[2][16 * PTS_LD];
  const int wave = threadIdx.x >> 5;
  const int lane = threadIdx.x & 31;
  const int hi = lane >> 4;
  const int l16 = lane & 15;

  const int tile = blockIdx.x * 4 + wave;
  const int b = tile >> 8;
  const int iBase = (tile & 255) << 4;

  const float* m = modes + (size_t)b * NPTS * NC;
  const float* mn = modeNorm + (size_t)b * NPTS;

  v2f a[4];
#pragma unroll
  for (int s = 0; s < 4; ++s) {
    const int k = 4 * s + 2 * hi;
    a[s] = *(const v2f*)(m + (size_t)(iBase + l16) * NC + k);
  }
  float mnr[8];
#pragma unroll
  for (int r = 0; r < 8; ++r) mnr[r] = mn[iBase + r + 8 * hi];

  int best[8];
#pragma unroll
  for (int r = 0; r < 8; ++r) best[r] = INT_MAX;

  if (wave == 0) {
    stage_tile(m, 0, mT[0], lane);
    tile_copy_wait();
  }
  __syncthreads();

  for (int jt = 0; jt < NPTS / 16; ++jt) {
    const int buf = jt & 1;
    const int jBase = jt << 4;
    if (wave == 0 && jt + 1 < NPTS / 16)
      stage_tile(m, jBase + 16, mT[buf ^ 1], lane);

    const float pnL = mn[jBase + l16];
    v8f acc = {};
#pragma unroll
    for (int s = 0; s < 4; ++s) {
      const int k = 4 * s + 2 * hi;
      const v2f bf = *(const v2f*)(&mT[buf][l16 * PTS_LD + k]);
      acc = __builtin_amdgcn_wmma_f32_16x16x4_f32(false, a[s], false, bf,
                                                  (short)0, acc, false, false);
    }
#pragma unroll
    for (int r = 0; r < 8; ++r) {
      const float d2 = mnr[r] + pnL - 2.0f * acc[r];
      const int cand = (d2 <= BW2) ? (jBase + l16) : INT_MAX;
      best[r] = min(best[r], cand);
    }

    if (wave == 0) tile_copy_wait();
    __syncthreads();
  }

#pragma unroll
  for (int r = 0; r < 8; ++r) {
    int v = best[r];
    v = min(v, __shfl_xor(v, 1, 32));
    v = min(v, __shfl_xor(v, 2, 32));
    v = min(v, __shfl_xor(v, 4, 32));
    v = min(v, __shfl_xor(v, 8, 32));
    if (l16 == 0) {
      const int row = iBase + r + 8 * hi;
      labels[b * NPTS + row] = (v == INT_MAX) ? 0 : v;
    }
  }
}

// ---------------------------------------------------------------------------
// Kernel 5/6/7: zero counts, majority vote accumulation, one-hot output.
// ---------------------------------------------------------------------------
__global__ __launch_bounds__(256) void k_zero(int* __restrict__ counts, int n) {
  const int tid = blockIdx.x * blockDim.x + threadIdx.x;
  if (tid < n) counts[tid] = 0;
}

__global__ __launch_bounds__(256) void k_votes(const int* __restrict__ labels,
                                               const int* __restrict__ segPred,
                                               int* __restrict__ counts) {
  const int tid = blockIdx.x * blockDim.x + threadIdx.x;  // 0..B*N-1
  const int b = tid >> 12;
  const int lab = labels[tid];
  atomicAdd(&counts[((size_t)b * NPTS + lab) * NK + segPred[tid]], 1);
}

__global__ __launch_bounds__(256) void k_out(const int* __restrict__ labels,
                                             const int* __restrict__ counts,
                                             float* __restrict__ out) {
  const int tid = blockIdx.x * blockDim.x + threadIdx.x;  // 0..B*N-1
  const int b = tid >> 12;
  const int n = tid & (NPTS - 1);
  const int lab = labels[tid];
  const int* c = &counts[((size_t)b * NPTS + lab) * NK];
  int bk = 0, bv = c[0];
#pragma unroll
  for (int k = 1; k < NK; ++k) {
    const int v = c[k];
    if (v > bv) { bv = v; bk = k; }  // first-max tie break (argmax semantics)
  }
#pragma unroll
  for (int k = 0; k < NK; ++k) {
    out[((size_t)(b * NK + k)) * NPTS + n] = (k == bk) ? 1.0f : 0.0f;
  }
}

// ---------------------------------------------------------------------------
extern "C" void kernel_launch(void* const* d_in, const int* in_sizes, int n_in,
                              void* d_out, int out_size, void* d_ws, size_t ws_size,
                              hipStream_t stream) {
  const float* emb = (const float*)d_in[0];  // [4,16,16,16,16] f32
  const float* seg = (const float*)d_in[1];  // [4,8,16,16,16]  f32
  float* out = (float*)d_out;                // [4,8,16,16,16]  f32

  // workspace layout (floats/ints), total ~3.75 MB
  float* wsf = (float*)d_ws;
  float* modesA = wsf;                          // B*N*C = 262144
  float* modesB = modesA + NB * NPTS * NC;      // 262144
  float* ptsRM = modesB + NB * NPTS * NC;       // 262144
  float* ptsNorm = ptsRM + NB * NPTS * NC;      // 16384
  float* modeNorm = ptsNorm + NB * NPTS;        // 16384
  int* labels = (int*)(modeNorm + NB * NPTS);   // 16384
  int* segPred = labels + NB * NPTS;            // 16384
  int* counts = segPred + NB * NPTS;            // B*N*K = 131072

  const int BN = NB * NPTS;  // 16384

  k_prep<<<BN / 256, 256, 0, stream>>>(emb, seg, ptsRM, modesA, ptsNorm, segPred);

  float* mIn = modesA;
  float* mOut = modesB;
  for (int it = 0; it < ITERS; ++it) {
    k_meanshift<<<256, 128, 0, stream>>>(mIn, mOut, ptsRM, ptsNorm);
    float* t = mIn; mIn = mOut; mOut = t;
  }
  // after an even number of swaps the final modes are in mIn (== modesA)

  k_norms<<<BN / 256, 256, 0, stream>>>(mIn, modeNorm);
  k_labels<<<256, 128, 0, stream>>>(mIn, modeNorm, labels);

  const int NCNT = NB * NPTS * NK;  // 131072
  k_zero<<<(NCNT + 255) / 256, 256, 0, stream>>>(counts, NCNT);
  k_votes<<<BN / 256, 256, 0, stream>>>(labels, segPred, counts);
  k_out<<<BN / 256, 256, 0, stream>>>(labels, counts, out);
}